// Emb_Layers_18279380811819
// MI455X (gfx1250) — compile-verified
//
#include <hip/hip_runtime.h>
#include <hip/hip_bf16.h>
#include <math.h>

// RGCN 2-layer forward for MI455X (gfx1250, wave32).
// Strategy: aggregate-then-transform. Scatter passes are the memory/atomic
// bound part; the per-(node,relation) 1x64 @ 64xDOUT transforms are done with
// V_WMMA_F32_16X16X4_F32 on 16-node tiles, weights staged in LDS.

typedef float v2f __attribute__((ext_vector_type(2)));
typedef float v8f __attribute__((ext_vector_type(8)));

#define RGCN_R 8
#define RGCN_DIN 64

// ---------------------------------------------------------------- zero fill
__global__ void rgcn_zero_kernel(float4* __restrict__ p, long n4) {
    long i = (long)blockIdx.x * blockDim.x + threadIdx.x;
    if (i < n4) p[i] = make_float4(0.f, 0.f, 0.f, 0.f);
}

// ------------------------------------------------------- per-segment counts
__global__ void rgcn_count_kernel(const int* __restrict__ dst,
                                  const int* __restrict__ et,
                                  float* __restrict__ cnt, int E) {
    int e = blockIdx.x * blockDim.x + threadIdx.x;
    if (e >= E) return;
    atomicAdd(&cnt[(size_t)dst[e] * RGCN_R + et[e]], 1.0f);
}

// --------------------------------------------- scatter-add x[src] into agg
// one wave (32 lanes) per edge; each lane handles 2 consecutive floats of the
// 64-float message (float2 gather + two f32 global atomics).
__global__ void rgcn_aggregate_kernel(const float* __restrict__ x,
                                      const int* __restrict__ src,
                                      const int* __restrict__ dst,
                                      const int* __restrict__ et,
                                      float* __restrict__ agg, int E) {
    long t = (long)blockIdx.x * blockDim.x + threadIdx.x;
    int e = (int)(t >> 5);
    if (e >= E) return;
    int lane = (int)(t & 31);
    int s = src[e];
    int d = dst[e];
    int r = et[e];
    const float2 v = *(const float2*)(x + (size_t)s * RGCN_DIN + lane * 2);
    float* p = agg + ((size_t)d * RGCN_R + r) * RGCN_DIN + lane * 2;
    atomicAdd(p + 0, v.x);
    atomicAdd(p + 1, v.y);
}

// ------------------------------------------------------ WMMA tile transform
// out[16xDOUT tile] = bias + xin(16x64) @ root + sum_r (agg[:,r,:]/cnt) @ W_r
// ACT: 0 = ReLU, 1 = sigmoid.  One wave per 16-node tile; 4 waves / WG share
// LDS-staged weights (R*64*DOUT + 64*DOUT floats).
template <int DOUT, int NT, int ACT>
__global__ __launch_bounds__(128) void rgcn_transform_kernel(
    const float* __restrict__ agg,   // [N, R, 64]
    const float* __restrict__ cnt,   // [N, R]
    const float* __restrict__ xin,   // [N, 64]
    const float* __restrict__ W,     // [R, 64, DOUT]
    const float* __restrict__ root,  // [64, DOUT]
    const float* __restrict__ bias,  // [DOUT]
    float* __restrict__ out,         // [N, DOUT]
    int nblocks) {
    extern __shared__ float lw[];  // [R*64*DOUT] weights | [64*DOUT] root
    const int nW = RGCN_R * RGCN_DIN * DOUT;
    const int nRoot = RGCN_DIN * DOUT;
    for (int i = threadIdx.x; i < nW; i += 128) lw[i] = W[i];
    for (int i = threadIdx.x; i < nRoot; i += 128) lw[nW + i] = root[i];
    __syncthreads();

    const int lane = threadIdx.x & 31;
    const int wave = threadIdx.x >> 5;
    const int m = lane & 15;            // A row / B,C column within tile
    const int khalf = (lane >> 4) * 2;  // K sub-offset: lanes<16 -> {0,1}, >=16 -> {2,3}

    for (int blk = blockIdx.x * 4 + wave; blk < nblocks; blk += gridDim.x * 4) {
        const int node = blk * 16 + m;

        v8f acc[NT];
#pragma unroll
        for (int t = 0; t < NT; ++t) {
            float b = bias[t * 16 + m];
            acc[t] = (v8f){b, b, b, b, b, b, b, b};
        }

        // ---- root term: xin(16x64) @ root(64xDOUT), scale 1
        const float* ar = xin + (size_t)node * RGCN_DIN;
        for (int k0 = 0; k0 < RGCN_DIN; k0 += 4) {
            v2f a;
            a.x = ar[k0 + khalf];
            a.y = ar[k0 + khalf + 1];
#pragma unroll
            for (int t = 0; t < NT; ++t) {
                v2f b;
                b.x = lw[nW + (k0 + khalf) * DOUT + t * 16 + m];
                b.y = lw[nW + (k0 + khalf + 1) * DOUT + t * 16 + m];
                acc[t] = __builtin_amdgcn_wmma_f32_16x16x4_f32(
                    false, a, false, b, (short)0, acc[t], false, false);
            }
        }

        // ---- relation terms: (agg[:,r,:]/max(cnt,1)) @ W_r
        for (int r = 0; r < RGCN_R; ++r) {
            const float sc = 1.0f / fmaxf(cnt[(size_t)node * RGCN_R + r], 1.0f);
            const float* ab = agg + ((size_t)node * RGCN_R + r) * RGCN_DIN;
            const int wbase = r * RGCN_DIN * DOUT;
            for (int k0 = 0; k0 < RGCN_DIN; k0 += 4) {
                v2f a;
                a.x = ab[k0 + khalf] * sc;
                a.y = ab[k0 + khalf + 1] * sc;
#pragma unroll
                for (int t = 0; t < NT; ++t) {
                    v2f b;
                    b.x = lw[wbase + (k0 + khalf) * DOUT + t * 16 + m];
                    b.y = lw[wbase + (k0 + khalf + 1) * DOUT + t * 16 + m];
                    acc[t] = __builtin_amdgcn_wmma_f32_16x16x4_f32(
                        false, a, false, b, (short)0, acc[t], false, false);
                }
            }
        }

        // ---- activation + store. C/D layout: M = j + 8*(lane>>4), N = lane&15
        const int rowbase = blk * 16 + 8 * (lane >> 4);
#pragma unroll
        for (int t = 0; t < NT; ++t) {
#pragma unroll
            for (int j = 0; j < 8; ++j) {
                float v = acc[t][j];
                v = (ACT == 0) ? fmaxf(v, 0.0f) : 1.0f / (1.0f + __expf(-v));
                out[(size_t)(rowbase + j) * DOUT + t * 16 + m] = v;
            }
        }
    }
}

extern "C" void kernel_launch(void* const* d_in, const int* in_sizes, int n_in,
                              void* d_out, int out_size, void* d_ws, size_t ws_size,
                              hipStream_t stream) {
    const float* emb   = (const float*)d_in[0];  // [N,64]
    const float* W1    = (const float*)d_in[1];  // [8,64,64]
    const float* root1 = (const float*)d_in[2];  // [64,64]
    const float* b1    = (const float*)d_in[3];  // [64]
    const float* W2    = (const float*)d_in[4];  // [8,64,32]
    const float* root2 = (const float*)d_in[5];  // [64,32]
    const float* b2    = (const float*)d_in[6];  // [32]
    const int* eidx    = (const int*)d_in[7];    // [2,E]
    const int* etype   = (const int*)d_in[8];    // [E]

    const int N = in_sizes[0] / RGCN_DIN;
    const int E = in_sizes[8];
    const int* src = eidx;
    const int* dst = eidx + E;

    float* ws  = (float*)d_ws;
    float* cnt = ws;                                    // N*R floats
    float* agg = ws + (size_t)N * RGCN_R;               // N*R*64 floats
    float* h   = agg + (size_t)N * RGCN_R * RGCN_DIN;   // N*64 floats
    float* out = (float*)d_out;                         // N*32 floats

    const size_t nZero1 = (size_t)N * RGCN_R * (1 + RGCN_DIN);  // cnt + agg
    const size_t nZeroA = (size_t)N * RGCN_R * RGCN_DIN;        // agg only
    const int nblocks = N / 16;

    // 1) zero cnt + agg
    {
        long n4 = (long)(nZero1 / 4);
        rgcn_zero_kernel<<<(int)((n4 + 255) / 256), 256, 0, stream>>>((float4*)ws, n4);
    }
    // 2) per-(dst,relation) edge counts
    rgcn_count_kernel<<<(E + 255) / 256, 256, 0, stream>>>(dst, etype, cnt, E);
    // 3) scatter-add layer-1 messages
    {
        long nt = (long)E * 32;
        rgcn_aggregate_kernel<<<(int)((nt + 255) / 256), 256, 0, stream>>>(
            emb, src, dst, etype, agg, E);
    }
    // 4) layer-1 WMMA transform -> h (ReLU)
    {
        const int smem = (RGCN_R + 1) * RGCN_DIN * 64 * sizeof(float);  // 147456 B
        rgcn_transform_kernel<64, 4, 0><<<512, 128, smem, stream>>>(
            agg, cnt, emb, W1, root1, b1, h, nblocks);
    }
    // 5) re-zero agg
    {
        long n4 = (long)(nZeroA / 4);
        rgcn_zero_kernel<<<(int)((n4 + 255) / 256), 256, 0, stream>>>((float4*)agg, n4);
    }
    // 6) scatter-add layer-2 messages (from h)
    {
        long nt = (long)E * 32;
        rgcn_aggregate_kernel<<<(int)((nt + 255) / 256), 256, 0, stream>>>(
            h, src, dst, etype, agg, E);
    }
    // 7) layer-2 WMMA transform -> out (sigmoid)
    {
        const int smem = (RGCN_R + 1) * RGCN_DIN * 32 * sizeof(float);  // 73728 B
        rgcn_transform_kernel<32, 2, 1><<<512, 128, smem, stream>>>(
            agg, cnt, h, W2, root2, b2, out, nblocks);
    }
}